// HYBLayer_pre_88072599371932
// MI455X (gfx1250) — compile-verified
//
#include <hip/hip_runtime.h>
#include <hip/hip_bf16.h>

#define D_IN 256
#define D_H  16
#define C_CH 6

typedef float v2f __attribute__((ext_vector_type(2)));
typedef float v4f __attribute__((ext_vector_type(4)));
typedef float v8f __attribute__((ext_vector_type(8)));

// -------------------------------------------------------------------------
// Zero the aggregation buffer with b128 stores (count is a multiple of 4).
// -------------------------------------------------------------------------
__global__ __launch_bounds__(256) void hyb_zero_kernel(v4f* __restrict__ p, long long n4) {
    long long t = (long long)blockIdx.x * blockDim.x + threadIdx.x;
    if (t < n4) p[t] = (v4f){0.0f, 0.0f, 0.0f, 0.0f};
}

// -------------------------------------------------------------------------
// h[c][n][0:16] = x[n][:] @ W[c][:][:]   via V_WMMA_F32_16X16X4_F32
// One wave (32 lanes) owns one 16-row tile and computes all 6 channels,
// reusing the A fragment 6x per K-step. K loop: 256/4 = 64 steps.
//
// A frag (16x4 f32): lane l -> m = l&15, kb = (l>>4)*2 ; holds A[m][kb..kb+1]
// B frag (4x16 f32): lane l -> n = l&15, kb = (l>>4)*2 ; holds B[kb..kb+1][n]
// D frag (16x16)   : lane l, vgpr j -> D[j + 8*(l>>4)][l&15]
//
// x is streamed once per tile -> non-temporal loads (preserve L2 for W/h/agg).
// W (96 KB) is reused by every wave -> regular temporal loads, L2-resident.
// -------------------------------------------------------------------------
__global__ __launch_bounds__(32) void hyb_gemm_wmma_kernel(
    const float* __restrict__ x,      // [n_nodes, 256]
    const float* __restrict__ W,      // [6, 256, 16]
    float* __restrict__ hbuf,         // [6, n_nodes, 16]
    int n_nodes)
{
    const int tile = blockIdx.x;            // 16-row tile index
    const int lane = threadIdx.x;           // 0..31
    const int m    = lane & 15;
    const int kb   = (lane >> 4) << 1;      // 0 or 2

    int row  = tile * 16 + m;
    int arow = row < n_nodes ? row : (n_nodes - 1);   // clamp loads; EXEC stays full

    v8f acc[C_CH] = {};

    const float* xrow = x + (size_t)arow * D_IN + kb;   // 8B aligned (kb even)

    for (int k = 0; k < D_IN; k += 4) {
        v2f a = __builtin_nontemporal_load((const v2f*)(xrow + k)); // global_load_b64 NT
#pragma unroll
        for (int c = 0; c < C_CH; ++c) {
            const float* wp = W + ((size_t)c * D_IN + (k + kb)) * D_H + m;
            v2f bf;
            bf.x = wp[0];
            bf.y = wp[D_H];
            acc[c] = __builtin_amdgcn_wmma_f32_16x16x4_f32(
                /*neg_a=*/false, a, /*neg_b=*/false, bf,
                /*c_mod=*/(short)0, acc[c],
                /*reuse_a=*/false, /*reuse_b=*/false);
        }
    }

    // Store D fragments: lane l holds rows j + 8*(l>>4), col l&15.
    // h is re-read by the SpMM phase -> regular (RT) stores, stays in L2.
    const int ncol  = lane & 15;
    const int mhalf = (lane >> 4) * 8;
#pragma unroll
    for (int c = 0; c < C_CH; ++c) {
        float* hc = hbuf + (size_t)c * n_nodes * D_H;
#pragma unroll
        for (int j = 0; j < 8; ++j) {
            int orow = tile * 16 + mhalf + j;
            if (orow < n_nodes)
                hc[(size_t)orow * D_H + ncol] = acc[c][j];
        }
    }
}

// -------------------------------------------------------------------------
// COO SpMM scatter: agg[c][rows[e]][f] += vals[e] * h[c][cols[e]][f]
// 4 consecutive lanes cooperate on one edge; each lane gathers a float4 of
// h (the 4 lanes cover one 64-byte segment -> coalesced b128 gathers) and
// issues 4 f32 atomicAdds into agg. Edge metadata is single-use stream ->
// non-temporal loads so the 230 MB of edge data does not evict the
// L2-resident h/agg working set (2 x 38.4 MB inside the 192 MB L2).
// -------------------------------------------------------------------------
__global__ __launch_bounds__(256) void hyb_spmm_kernel(
    const float* __restrict__ vals,   // [6, n_edges]
    const int*   __restrict__ rows,   // [6, n_edges]
    const int*   __restrict__ cols,   // [6, n_edges]
    const float* __restrict__ hbuf,   // [6, n_nodes, 16]
    float*       __restrict__ agg,    // [6, n_nodes, 16]
    int n_nodes, int n_edges)
{
    const int c = blockIdx.y;
    long long t = (long long)blockIdx.x * blockDim.x + threadIdx.x;
    const int q = (int)(t & 3);            // quarter index: features q*4 .. q*4+3
    long long e = t >> 2;
    if (e >= n_edges) return;

    size_t ebase = (size_t)c * n_edges + (size_t)e;
    float  v   = __builtin_nontemporal_load(vals + ebase);
    int    r   = __builtin_nontemporal_load(rows + ebase);
    int    col = __builtin_nontemporal_load(cols + ebase);

    const v4f* hc4 = (const v4f*)(hbuf + ((size_t)c * n_nodes + (size_t)col) * D_H);
    v4f hv = hc4[q];                       // global_load_b128, L2 hit expected

    float* ap = agg + ((size_t)c * n_nodes + (size_t)r) * D_H + q * 4;
    atomicAdd(ap + 0, v * hv.x);
    atomicAdd(ap + 1, v * hv.y);
    atomicAdd(ap + 2, v * hv.z);
    atomicAdd(ap + 3, v * hv.w);
}

// -------------------------------------------------------------------------
// out[n][c*16+h] = relu(agg[c][n][h] + b[c][h])   ([C,N,16] -> [N,96])
// Vectorized by 4 (4 | 16 so a float4 never crosses a channel boundary).
// out is write-once -> non-temporal stores.
// -------------------------------------------------------------------------
__global__ __launch_bounds__(256) void hyb_bias_relu_kernel(
    const float* __restrict__ agg,    // [6, n_nodes, 16]
    const float* __restrict__ b,      // [6, 16]
    float*       __restrict__ out,    // [n_nodes, 96]
    int n_nodes)
{
    long long t4 = (long long)blockIdx.x * blockDim.x + threadIdx.x;
    long long total4 = (long long)n_nodes * (C_CH * D_H) / 4;
    if (t4 >= total4) return;

    long long flat = t4 * 4;
    int fc = (int)(flat % (C_CH * D_H));   // multiple of 4
    long long n = flat / (C_CH * D_H);
    int c = fc >> 4;
    int h = fc & 15;                       // multiple of 4

    v4f av = *(const v4f*)(agg + ((size_t)c * n_nodes + (size_t)n) * D_H + h);
    v4f bv = *(const v4f*)(b + c * D_H + h);
    v4f r;
    r.x = av.x + bv.x; r.y = av.y + bv.y; r.z = av.z + bv.z; r.w = av.w + bv.w;
    r.x = r.x > 0.0f ? r.x : 0.0f;
    r.y = r.y > 0.0f ? r.y : 0.0f;
    r.z = r.z > 0.0f ? r.z : 0.0f;
    r.w = r.w > 0.0f ? r.w : 0.0f;
    __builtin_nontemporal_store(r, (v4f*)(out + flat));
}

// -------------------------------------------------------------------------
// Launcher
// Inputs: x [N,256] f32 | W [6,256,16] f32 | b [6,16] f32 |
//         edge_vals [6,E] f32 | edge_rows [6,E] i32 | edge_cols [6,E] i32
// Output: [N, 96] f32
// Workspace: h   @ ws + 0              (6*N*16 f32 = 38.4 MB)
//            agg @ ws + 6*N*16*4       (6*N*16 f32 = 38.4 MB)
// -------------------------------------------------------------------------
extern "C" void kernel_launch(void* const* d_in, const int* in_sizes, int n_in,
                              void* d_out, int out_size, void* d_ws, size_t ws_size,
                              hipStream_t stream) {
    const float* x    = (const float*)d_in[0];
    const float* W    = (const float*)d_in[1];
    const float* b    = (const float*)d_in[2];
    const float* vals = (const float*)d_in[3];
    const int*   rows = (const int*)d_in[4];
    const int*   cols = (const int*)d_in[5];
    float*       out  = (float*)d_out;

    const int n_nodes = in_sizes[0] / D_IN;
    const int n_edges = in_sizes[3] / C_CH;

    float* hbuf = (float*)d_ws;
    float* agg  = hbuf + (size_t)C_CH * n_nodes * D_H;

    // 1) zero agg (b128 stores)
    {
        long long n4 = (long long)C_CH * n_nodes * D_H / 4;
        int blocks = (int)((n4 + 255) / 256);
        hyb_zero_kernel<<<blocks, 256, 0, stream>>>((v4f*)agg, n4);
    }
    // 2) WMMA GEMM: h = x @ W (all 6 channels per wave)
    {
        int tiles = (n_nodes + 15) / 16;   // 6250 (exact)
        hyb_gemm_wmma_kernel<<<tiles, 32, 0, stream>>>(x, W, hbuf, n_nodes);
    }
    // 3) SpMM scatter-add (4 lanes per edge, b128 gathers, NT edge streams)
    {
        long long threads = (long long)n_edges * 4;
        int blocks = (int)((threads + 255) / 256);
        dim3 grid(blocks, C_CH, 1);
        hyb_spmm_kernel<<<grid, 256, 0, stream>>>(vals, rows, cols, hbuf, agg,
                                                  n_nodes, n_edges);
    }
    // 4) bias + relu + channel-concat transpose (b128, NT stores)
    {
        long long n4 = (long long)n_nodes * (C_CH * D_H) / 4;
        int blocks = (int)((n4 + 255) / 256);
        hyb_bias_relu_kernel<<<blocks, 256, 0, stream>>>(agg, b, out, n_nodes);
    }
}